// LSTM_72705206387295
// MI455X (gfx1250) — compile-verified
//
#include <hip/hip_runtime.h>

// ---------------------------------------------------------------------------
// Fused LSTM for MI455X (gfx1250, wave32).  B=4096, T=2048, I=13, H=3, O=1.
// - One wave owns 16 batch rows for the whole T=2048 recurrence.
// - Per step: g(16x12) = [x_t | h](16x16) * [W_ih^T ; W_hh^T](16x16) + bias
//   as a chain of 4x V_WMMA_F32_16X16X4_F32 (full fp32, matches reference).
// - x is streamed with CDNA5 async copies (GLOBAL_LOAD_ASYNC_TO_LDS_B128),
//   double-buffered in LDS and synchronized with s_wait_asynccnt, so staging
//   overlaps compute. global_prefetch warms the stage after next.
// - Activations use the gfx1250 hardware v_tanh_f32 transcendental.
// ---------------------------------------------------------------------------

typedef float v8f __attribute__((ext_vector_type(8)));
typedef float v2f __attribute__((ext_vector_type(2)));
typedef __attribute__((address_space(3))) float lds_f32;

constexpr int Bn = 4096;
constexpr int Tn = 2048;
constexpr int In = 13;          // input features
constexpr int Hn = 3;           // hidden
constexpr int G4 = 12;          // 4*H gates

constexpr int TC      = 16;               // timesteps per LDS stage
constexpr int SEG_DW  = TC * In;          // 208 dwords per batch-row segment
constexpr int SEG_PAD = 212;              // padded stride: 16B-aligned, bank-spread
constexpr int CHUNKS  = SEG_DW / 4;       // 52 float4 chunks per segment
constexpr int STAGES  = Tn / TC;          // 128
constexpr int ISSUES  = 16 * CHUNKS / 32; // 26 async b128 issues per lane per stage
static_assert(ISSUES == 26, "wait-count literal below assumes 26 issues/stage");

#define USE_ASYNC 1

#if USE_ASYNC
#define ASYNC_FENCE(nlit) asm volatile("s_wait_asynccnt " nlit ::: "memory")
#else
#define ASYNC_FENCE(nlit) asm volatile("" ::: "memory")
#endif

#if __has_builtin(__builtin_amdgcn_wmma_f32_16x16x4_f32)
#define HAVE_WMMA_F32X4 1
#else
#define HAVE_WMMA_F32X4 0
#endif

#if __has_builtin(__builtin_amdgcn_tanhf)
__device__ __forceinline__ float tanh_f(float v) { return __builtin_amdgcn_tanhf(v); }
#else
__device__ __forceinline__ float tanh_f(float v) {
    return 2.0f * __builtin_amdgcn_rcpf(1.0f + __expf(-2.0f * v)) - 1.0f;
}
#endif
__device__ __forceinline__ float sig_f(float v) {
    return __builtin_fmaf(0.5f, tanh_f(0.5f * v), 0.5f);   // sigmoid via tanh
}

__global__ __launch_bounds__(32, 1)
void lstm_wmma_fused(const float* __restrict__ x,
                     const float* __restrict__ h0,
                     const float* __restrict__ c0,
                     const float* __restrict__ W_ih,
                     const float* __restrict__ W_hh,
                     const float* __restrict__ b_ih,
                     const float* __restrict__ b_hh,
                     const float* __restrict__ W_fc,
                     const float* __restrict__ b_fc,
                     float* __restrict__ out)
{
    __shared__ float xbuf[2][16 * SEG_PAD];  // double-buffered x stage, 2x13.6 KB
    __shared__ float gbuf[16 * 16];          // WMMA D-tile exchange (row stride 16)
    __shared__ float hbuf[16 * Hn];          // hidden state, readable in A-layout

    const int lane = threadIdx.x;            // 0..31 (wave32)
    const int m    = lane & 15;              // batch row within tile / B-column n
    const int half = lane >> 4;              // 0: lanes 0-15, 1: lanes 16-31
    const int b0   = blockIdx.x << 4;
    const int bg   = b0 + m;                 // this lane's global batch row

    // ---- B operand: [W_ih^T ; W_hh^T] as 16x16, column n = m (n>=12 -> 0) ----
    // 16x4 f32 operand layout: reg0 = K(2*half), reg1 = K(2*half+1) per chunk.
    auto wval = [&](int k) -> float {
        if (m >= G4) return 0.0f;                    // zero-padded columns 12..15
        if (k < In)  return W_ih[m * In + k];        // W_ih is [12,13]
        int j = k - In;                              // 0..2
        return (j < Hn) ? W_hh[m * Hn + j] : 0.0f;   // W_hh is [12,3]
    };
    v2f bm[4];
#pragma unroll
    for (int q = 0; q < 4; ++q) {
        const int k0 = 4 * q + 2 * half;
        bm[q].x = wval(k0);
        bm[q].y = wval(k0 + 1);
    }

    // Bias tile: every row gets bias[n] in column n.
    float bsum = 0.0f;
    if (m < G4) bsum = b_ih[m] + b_hh[m];
    v8f cbias;
#pragma unroll
    for (int r = 0; r < 8; ++r) cbias[r] = bsum;

    // ---- state init (both halves redundantly track the same 16 rows) ----
    float c[Hn];
#pragma unroll
    for (int j = 0; j < Hn; ++j) {
        c[j] = c0[bg * Hn + j];
        hbuf[m * Hn + j] = h0[bg * Hn + j];  // same value from both halves: benign
    }

    // ---- async stage issue: 26 b128 copies/lane, global -> LDS, coalesced ----
    auto issue_stage = [&](int s, int buf) {
#pragma unroll 1
        for (int it = 0; it < ISSUES; ++it) {
            const int cid = it * 32 + lane;
            const int seg = cid / CHUNKS;
            const int off = cid - seg * CHUNKS;                 // 16B units
            const char* gp = (const char*)x
                + ((size_t)(b0 + seg) * Tn * In) * sizeof(float)
                + (size_t)s * (TC * In * sizeof(float))
                + (size_t)off * 16;
            float* lp = &xbuf[buf][seg * SEG_PAD + off * 4];
#if USE_ASYNC
            asm volatile("global_load_async_to_lds_b128 %0, %1, off"
                         :: "v"((lds_f32*)lp), "v"(gp) : "memory");
#else
            *(float4*)lp = *(const float4*)gp;
#endif
            if (s + 1 < STAGES)
                __builtin_prefetch(gp + TC * In * sizeof(float), 0, 3); // next stage
        }
    };

    issue_stage(0, 0);

#pragma unroll 1
    for (int s = 0; s < STAGES; ++s) {
        if (s + 1 < STAGES) {
            issue_stage(s + 1, (s + 1) & 1);   // overlap next stage with compute
            ASYNC_FENCE("26");                 // oldest 26 (stage s) have landed
        } else {
            ASYNC_FENCE("0");
        }
        const float* xs = xbuf[s & 1];

        // ---- TC recurrence steps out of LDS ----
#pragma unroll 1
        for (int tl = 0; tl < TC; ++tl) {
            const float* xl = &xs[m * SEG_PAD + tl * In];
            const int kb = 2 * half;

            v2f a0, a1, a2, a3;
            a0.x = xl[kb + 0]; a0.y = xl[kb + 1];
            a1.x = xl[kb + 4]; a1.y = xl[kb + 5];
            a2.x = xl[kb + 8]; a2.y = xl[kb + 9];
            // K=12..15: x[12] then h[0..2] (address select keeps control flow uniform)
            const float* a3xp = half ? &hbuf[m * Hn + 1] : &xl[12];
            a3.x = *a3xp;
            a3.y = hbuf[m * Hn + (half ? 2 : 0)];

#if HAVE_WMMA_F32X4
            v8f acc = cbias;
            acc = __builtin_amdgcn_wmma_f32_16x16x4_f32(false, a0, false, bm[0], (short)0, acc, false, false);
            acc = __builtin_amdgcn_wmma_f32_16x16x4_f32(false, a1, false, bm[1], (short)0, acc, false, false);
            acc = __builtin_amdgcn_wmma_f32_16x16x4_f32(false, a2, false, bm[2], (short)0, acc, false, false);
            acc = __builtin_amdgcn_wmma_f32_16x16x4_f32(false, a3, false, bm[3], (short)0, acc, false, false);
            // D-tile: lane holds column n=m for rows r(+8*half); scatter to LDS.
#pragma unroll
            for (int r = 0; r < 8; ++r)
                gbuf[(r + 8 * half) * 16 + m] = acc[r];
#else
            // Scalar fallback keeps the same dataflow.
#pragma unroll
            for (int n = 0; n < G4; ++n) {
                float g = b_ih[n] + b_hh[n];
#pragma unroll
                for (int k = 0; k < In; ++k) g += xl[k] * W_ih[n * In + k];
#pragma unroll
                for (int j = 0; j < Hn; ++j) g += hbuf[m * Hn + j] * W_hh[n * Hn + j];
                gbuf[m * 16 + n] = g;
            }
#endif
            // Gather this lane's row of gates (12 contiguous dwords -> b128 loads).
            float g[G4];
#pragma unroll
            for (int n = 0; n < G4; ++n) g[n] = gbuf[m * 16 + n];

            // LSTM cell update (torch gate order i,f,g,o). Both halves redundant.
#pragma unroll
            for (int j = 0; j < Hn; ++j) {
                const float ig = sig_f(g[j]);
                const float fg = sig_f(g[3 + j]);
                const float gg = tanh_f(g[6 + j]);
                const float og = sig_f(g[9 + j]);
                c[j] = fg * c[j] + ig * gg;
                hbuf[m * Hn + j] = og * tanh_f(c[j]);
            }
        }
    }

    // ---- epilogue: out = hN @ W_fc^T + b_fc ; emit (out, hN, cN) ----
    if (lane < 16) {
        float hv[Hn];
#pragma unroll
        for (int j = 0; j < Hn; ++j) hv[j] = hbuf[m * Hn + j];
        float o = b_fc[0];
#pragma unroll
        for (int j = 0; j < Hn; ++j) o += hv[j] * W_fc[j];
        out[bg] = o;
        float* hN = out + Bn;
        float* cN = out + Bn + Bn * Hn;
#pragma unroll
        for (int j = 0; j < Hn; ++j) {
            hN[bg * Hn + j] = hv[j];
            cN[bg * Hn + j] = c[j];
        }
    }
}

extern "C" void kernel_launch(void* const* d_in, const int* in_sizes, int n_in,
                              void* d_out, int out_size, void* d_ws, size_t ws_size,
                              hipStream_t stream)
{
    (void)in_sizes; (void)n_in; (void)out_size; (void)d_ws; (void)ws_size;
    const float* x    = (const float*)d_in[0];
    const float* h0   = (const float*)d_in[1];
    const float* c0   = (const float*)d_in[2];
    const float* W_ih = (const float*)d_in[3];
    const float* W_hh = (const float*)d_in[4];
    const float* b_ih = (const float*)d_in[5];
    const float* b_hh = (const float*)d_in[6];
    const float* W_fc = (const float*)d_in[7];
    const float* b_fc = (const float*)d_in[8];

    lstm_wmma_fused<<<Bn / 16, 32, 0, stream>>>(
        x, h0, c0, W_ih, W_hh, b_ih, b_hh, W_fc, b_fc, (float*)d_out);
}